// GCNModel_20246475833484
// MI455X (gfx1250) — compile-verified
//
#include <hip/hip_runtime.h>

typedef __attribute__((ext_vector_type(2))) float v2f;
typedef __attribute__((ext_vector_type(8))) float v8f;

#define G_    8
#define N_    10000
#define E_    160000
#define DIN_  300
#define DH_   256
#define DOUT_ 128
#define EPS_  1e-5f

// ---------------------------------------------------------------------------
// 1) fill: agg <- 0, deg <- 1.0 (self-loop)
// ---------------------------------------------------------------------------
__global__ void __launch_bounds__(256)
fill_kernel(float* __restrict__ agg, long agg_n, float* __restrict__ deg, long deg_n) {
    long i = (long)blockIdx.x * blockDim.x + threadIdx.x;
    if (i < agg_n) agg[i] = 0.0f;
    if (i < deg_n) deg[i] = 1.0f;
}

// ---------------------------------------------------------------------------
// 2) in-degree accumulation over all (g, e)
// ---------------------------------------------------------------------------
__global__ void __launch_bounds__(256)
degree_kernel(const int* __restrict__ ei, float* __restrict__ deg) {
    long i = (long)blockIdx.x * blockDim.x + threadIdx.x;
    if (i >= (long)G_ * E_) return;
    int g = (int)(i / E_);
    int e = (int)(i - (long)g * E_);
    int dst = ei[(long)g * 2 * E_ + E_ + e];
    atomicAdd(&deg[g * N_ + dst], 1.0f);
}

// ---------------------------------------------------------------------------
// 3) deg -> rsqrt(deg) in place
// ---------------------------------------------------------------------------
__global__ void __launch_bounds__(256)
rsqrt_kernel(float* __restrict__ deg) {
    int i = blockIdx.x * blockDim.x + threadIdx.x;
    if (i < G_ * N_) deg[i] = rsqrtf(deg[i]);
}

// ---------------------------------------------------------------------------
// 4) fp32 WMMA GEMM: D[g] = A[g] @ W   (A: [N_,K] row-major, W: [K,F] row-major)
//    One wave32 per 16x16 output tile, K-loop in steps of 4 using
//    V_WMMA_F32_16X16X4_F32.
//    A 16x4 layout: lanes 0-15 hold (K=0,K=1), lanes 16-31 hold (K=2,K=3).
//    B 4x16 layout mirrors A; C/D: VGPR r -> row (r + 8*(lane>>4)), col lane&15.
// ---------------------------------------------------------------------------
template <int K, int F>
__global__ void __launch_bounds__(256)
gemm_wmma_kernel(const float* __restrict__ A, const float* __restrict__ W,
                 float* __restrict__ D, int total_tiles) {
    const int wave = blockIdx.x * (blockDim.x >> 5) + (threadIdx.x >> 5);
    if (wave >= total_tiles) return;
    const int lane = threadIdx.x & 31;

    constexpr int TILES_N = F / 16;
    constexpr int TILES_M = N_ / 16;

    int g   = wave / (TILES_M * TILES_N);
    int rem = wave - g * (TILES_M * TILES_N);
    int tm  = rem / TILES_N;
    int tn  = rem - tm * TILES_N;

    const float* Ag = A + (long)g * N_ * K;
    float*       Dg = D + (long)g * N_ * F;

    const int half = lane >> 4;    // 0: K pair (0,1), 1: K pair (2,3)
    const int lid  = lane & 15;

    const int arow = tm * 16 + lid;
    const int bcol = tn * 16 + lid;

    const float* aptr = Ag + (long)arow * K + 2 * half;   // 8B aligned (K % 4 == 0)
    const float* bptr = W + (long)(2 * half) * F + bcol;

    v8f acc = {};
    for (int k = 0; k < K; k += 4) {
        v2f a = *(const v2f*)(aptr + k);
        v2f b;
        b.x = bptr[(long)k * F];
        b.y = bptr[(long)(k + 1) * F];
        __builtin_prefetch(aptr + k + 64, 0, 0);   // global_prefetch_b8 on A stream
        acc = __builtin_amdgcn_wmma_f32_16x16x4_f32(
            /*neg_a=*/false, a, /*neg_b=*/false, b,
            /*c_mod=*/(short)0, acc, /*reuse_a=*/false, /*reuse_b=*/false);
    }

    float* drow = Dg + (long)(tm * 16 + 8 * half) * F + bcol;
#pragma unroll
    for (int r = 0; r < 8; ++r)
        drow[(long)r * F] = acc[r];
}

// ---------------------------------------------------------------------------
// 5) edge aggregation: one wave per (g, e); F/32 atomic f32 adds per lane
//    agg[dst] += h[src] * dinv[src]*dinv[dst]
// ---------------------------------------------------------------------------
template <int F>
__global__ void __launch_bounds__(256)
agg_kernel(const int* __restrict__ ei, const float* __restrict__ h,
           const float* __restrict__ dinv, float* __restrict__ agg) {
    const long wave = (long)blockIdx.x * (blockDim.x >> 5) + (threadIdx.x >> 5);
    if (wave >= (long)G_ * E_) return;
    const int lane = threadIdx.x & 31;
    int g = (int)(wave / E_);
    int e = (int)(wave - (long)g * E_);
    const int* eg = ei + (long)g * 2 * E_;
    int src = eg[e];
    int dst = eg[E_ + e];
    float coef = dinv[g * N_ + src] * dinv[g * N_ + dst];
    const float* hs = h + ((long)g * N_ + src) * F;
    float*       ad = agg + ((long)g * N_ + dst) * F;
#pragma unroll
    for (int i = 0; i < F / 32; ++i) {
        int f = lane + 32 * i;
        atomicAdd(&ad[f], hs[f] * coef);
    }
}

// ---------------------------------------------------------------------------
// 6) fused self-loop + bias (+ReLU) + LayerNorm; one wave32 per node.
//    Optionally re-zeroes its agg row so the buffer can be reused by layer 2.
// ---------------------------------------------------------------------------
template <int F, bool RELU>
__global__ void __launch_bounds__(256)
post_kernel(const float* __restrict__ agg, const float* __restrict__ h,
            const float* __restrict__ dinv, const float* __restrict__ bias,
            const float* __restrict__ gamma, const float* __restrict__ beta,
            float* __restrict__ out, float* __restrict__ agg_zero) {
    const int wave = blockIdx.x * (blockDim.x >> 5) + (threadIdx.x >> 5);
    if (wave >= G_ * N_) return;
    const int lane = threadIdx.x & 31;
    const long row = (long)wave * F;
    const float di  = dinv[wave];
    const float di2 = di * di;

    constexpr int V = F / 32;
    float v[V];
    float s = 0.0f;
#pragma unroll
    for (int i = 0; i < V; ++i) {
        int f = lane + 32 * i;
        float x = agg[row + f] + h[row + f] * di2 + bias[f];
        if (RELU) x = fmaxf(x, 0.0f);
        v[i] = x;
        s += x;
    }
#pragma unroll
    for (int o = 16; o > 0; o >>= 1) s += __shfl_xor(s, o, 32);
    const float mu = s * (1.0f / F);

    float vs = 0.0f;
#pragma unroll
    for (int i = 0; i < V; ++i) { float d = v[i] - mu; vs += d * d; }
#pragma unroll
    for (int o = 16; o > 0; o >>= 1) vs += __shfl_xor(vs, o, 32);
    const float inv = rsqrtf(vs * (1.0f / F) + EPS_);

#pragma unroll
    for (int i = 0; i < V; ++i) {
        int f = lane + 32 * i;
        out[row + f] = (v[i] - mu) * inv * gamma[f] + beta[f];
        if (agg_zero) agg_zero[row + f] = 0.0f;
    }
}

// ---------------------------------------------------------------------------
// launch
// ---------------------------------------------------------------------------
extern "C" void kernel_launch(void* const* d_in, const int* in_sizes, int n_in,
                              void* d_out, int out_size, void* d_ws, size_t ws_size,
                              hipStream_t stream) {
    const float* x   = (const float*)d_in[0];
    const int*   ei  = (const int*)d_in[1];
    const float* W1  = (const float*)d_in[2];
    const float* b1  = (const float*)d_in[3];
    const float* g1  = (const float*)d_in[4];
    const float* be1 = (const float*)d_in[5];
    const float* W2  = (const float*)d_in[6];
    const float* b2  = (const float*)d_in[7];
    const float* g2  = (const float*)d_in[8];
    const float* be2 = (const float*)d_in[9];
    float* out = (float*)d_out;

    // workspace carve (floats): agg[G,N,256] | h1[G,N,256] | h2[G,N,128] | dinv[G,N]
    float* ws   = (float*)d_ws;
    float* agg  = ws;
    float* h1   = agg + (size_t)G_ * N_ * DH_;
    float* h2   = h1 + (size_t)G_ * N_ * DH_;
    float* dinv = h2 + (size_t)G_ * N_ * DOUT_;

    const long agg_n = (long)G_ * N_ * DH_;     // 20,480,000
    const long deg_n = (long)G_ * N_;           // 80,000

    // 1) init
    {
        long threads = agg_n;
        int blocks = (int)((threads + 255) / 256);
        fill_kernel<<<blocks, 256, 0, stream>>>(agg, agg_n, dinv, deg_n);
    }
    // 2) degrees
    {
        long threads = (long)G_ * E_;
        int blocks = (int)((threads + 255) / 256);
        degree_kernel<<<blocks, 256, 0, stream>>>(ei, dinv);
    }
    // 3) dinv = rsqrt(deg)
    rsqrt_kernel<<<(G_ * N_ + 255) / 256, 256, 0, stream>>>(dinv);

    // ----- layer 1 -----
    {
        int tiles = G_ * (N_ / 16) * (DH_ / 16);            // 80,000
        gemm_wmma_kernel<DIN_, DH_><<<(tiles + 7) / 8, 256, 0, stream>>>(x, W1, h1, tiles);
    }
    {
        long waves = (long)G_ * E_;                         // 1,280,000
        agg_kernel<DH_><<<(int)((waves + 7) / 8), 256, 0, stream>>>(ei, h1, dinv, agg);
    }
    {
        int waves = G_ * N_;                                // 80,000
        // writes normalized result into h1 in place; zeroes agg rows for reuse
        post_kernel<DH_, true><<<(waves + 7) / 8, 256, 0, stream>>>(
            agg, h1, dinv, b1, g1, be1, h1, agg);
    }

    // ----- layer 2 -----
    {
        int tiles = G_ * (N_ / 16) * (DOUT_ / 16);          // 40,000
        gemm_wmma_kernel<DH_, DOUT_><<<(tiles + 7) / 8, 256, 0, stream>>>(h1, W2, h2, tiles);
    }
    {
        long waves = (long)G_ * E_;
        agg_kernel<DOUT_><<<(int)((waves + 7) / 8), 256, 0, stream>>>(ei, h2, dinv, agg);
    }
    {
        int waves = G_ * N_;
        post_kernel<DOUT_, false><<<(waves + 7) / 8, 256, 0, stream>>>(
            agg, h2, dinv, b2, g2, be2, out, nullptr);
    }
    (void)in_sizes; (void)n_in; (void)out_size; (void)ws_size;
}